// PipelinedMoEBlock_12249246728962
// MI455X (gfx1250) — compile-verified
//
#include <hip/hip_runtime.h>
#include <hip/hip_bf16.h>
#include <math.h>

typedef __attribute__((ext_vector_type(2))) float v2f;
typedef __attribute__((ext_vector_type(8))) float v8f;
typedef __attribute__((ext_vector_type(4))) unsigned int v4u;
typedef __attribute__((ext_vector_type(8))) int v8i;
typedef __attribute__((ext_vector_type(4))) int v4i;

static __device__ __forceinline__ v8f wmma4(v2f a, v2f b, v8f c) {
  // D = A(16x4 f32) * B(4x16 f32) + C(16x16 f32)
  return __builtin_amdgcn_wmma_f32_16x16x4_f32(false, a, false, b, (short)0, c, false, false);
}
static __device__ __forceinline__ v8f v8zero() {
  v8f z;
#pragma unroll
  for (int i = 0; i < 8; ++i) z[i] = 0.0f;
  return z;
}
static __device__ __forceinline__ float gelu_tanh(float x) {
  const float c0 = 0.7978845608028654f; // sqrt(2/pi)
  float u = c0 * (x + 0.044715f * x * x * x);
  return 0.5f * x * (1.0f + tanhf(u));
}
// LDS byte offset of a __shared__ object (addrspacecast generic -> AS3, 32-bit)
static __device__ __forceinline__ unsigned int lds_offset_of(const void* p) {
  return (unsigned int)(uintptr_t)(__attribute__((address_space(3))) const void*)p;
}

// ---------------- LayerNorm: one block per token row ----------------
__global__ __launch_bounds__(256) void ln_kernel(const float* __restrict__ x,
                                                 const float* __restrict__ g,
                                                 const float* __restrict__ b,
                                                 float* __restrict__ out, int D) {
  const int t = blockIdx.x, tid = threadIdx.x;
  const float* xr = x + (size_t)t * D;
  float s = 0.f, s2 = 0.f;
  for (int d = tid; d < D; d += 256) { float v = xr[d]; s += v; s2 += v * v; }
  __shared__ float r1[256], r2[256];
  r1[tid] = s; r2[tid] = s2;
  __syncthreads();
  for (int st = 128; st > 0; st >>= 1) {
    if (tid < st) { r1[tid] += r1[tid + st]; r2[tid] += r2[tid + st]; }
    __syncthreads();
  }
  const float mu = r1[0] / (float)D;
  const float var = r2[0] / (float)D - mu * mu;
  const float rstd = rsqrtf(var + 1e-5f);
  float* orow = out + (size_t)t * D;
  for (int d = tid; d < D; d += 256) orow[d] = (xr[d] - mu) * rstd * g[d] + b[d];
}

// ---------------- Generic WMMA GEMM: 64x64 tile / 256-thread block ----------------
// MODE 1: scatter to [B,H,S,Dh] layout (QKV).  MODE 2: out = res + A@B (Wo residual).
template <int MODE>
__global__ __launch_bounds__(256) void gemm64_kernel(const float* __restrict__ A,
                                                     const float* __restrict__ B,
                                                     const float* __restrict__ res,
                                                     float* __restrict__ out,
                                                     int M, int N, int K,
                                                     int S, int H, int Dh) {
  const int tid = threadIdx.x;
  const int lane = tid & 31, wave = tid >> 5;
  const int half = lane >> 4, l16 = lane & 15;
  const int sr = wave >> 1;            // A 16-row block of this wave (0..3)
  const int scb = (wave & 1) * 2;      // first of two 16-col subtiles (0 or 2)
  const int m0 = blockIdx.y * 64, n0 = blockIdx.x * 64;

  __shared__ float As[64 * 17];        // 64 rows x 16 k (pad 17 -> no bank conflicts)
  __shared__ float Bs[16 * 68];        // 16 k x 64 n (pad 68)

  v8f acc[2]; acc[0] = v8zero(); acc[1] = v8zero();

  const int ar = tid >> 2, ac = (tid & 3) * 4;
  const int br = tid >> 4, bc = (tid & 15) * 4;

  for (int kc = 0; kc < K; kc += 16) {
    float4 av = *(const float4*)(A + (size_t)(m0 + ar) * K + kc + ac);
    float4 bv = *(const float4*)(B + (size_t)(kc + br) * N + n0 + bc);
    As[ar * 17 + ac + 0] = av.x; As[ar * 17 + ac + 1] = av.y;
    As[ar * 17 + ac + 2] = av.z; As[ar * 17 + ac + 3] = av.w;
    Bs[br * 68 + bc + 0] = bv.x; Bs[br * 68 + bc + 1] = bv.y;
    Bs[br * 68 + bc + 2] = bv.z; Bs[br * 68 + bc + 3] = bv.w;
    __syncthreads();
#pragma unroll
    for (int kk = 0; kk < 16; kk += 4) {
      v2f a;  a[0]  = As[(sr * 16 + l16) * 17 + kk + half * 2];
              a[1]  = As[(sr * 16 + l16) * 17 + kk + half * 2 + 1];
      v2f b0; b0[0] = Bs[(kk + half * 2) * 68 + scb * 16 + l16];
              b0[1] = Bs[(kk + half * 2 + 1) * 68 + scb * 16 + l16];
      v2f b1; b1[0] = Bs[(kk + half * 2) * 68 + (scb + 1) * 16 + l16];
              b1[1] = Bs[(kk + half * 2 + 1) * 68 + (scb + 1) * 16 + l16];
      acc[0] = wmma4(a, b0, acc[0]);
      acc[1] = wmma4(a, b1, acc[1]);
    }
    __syncthreads();
  }
#pragma unroll
  for (int i = 0; i < 2; ++i) {
#pragma unroll
    for (int r = 0; r < 8; ++r) {
      int gm = m0 + sr * 16 + r + 8 * half;
      int gn = n0 + (scb + i) * 16 + l16;
      float val = acc[i][r];
      if (MODE == 1) {
        int bb = gm / S, ss = gm % S;
        int hh = gn / Dh, dd = gn % Dh;
        out[((size_t)(bb * H + hh) * S + ss) * Dh + dd] = val;
      } else {
        size_t idx = (size_t)gm * N + gn;
        out[idx] = res[idx] + val;
      }
    }
  }
}

// ---------------- Flash-style attention: one wave per 16-row q tile ----------------
__global__ __launch_bounds__(32) void attn_kernel(const float* __restrict__ q,
                                                  const float* __restrict__ k,
                                                  const float* __restrict__ v,
                                                  float* __restrict__ ctx,
                                                  int S, int H, int D) {
  const int qt = blockIdx.x, hh = blockIdx.y, b = blockIdx.z;
  const int lane = threadIdx.x & 31;
  const int half = lane >> 4, l16 = lane & 15;
  const float* Qb = q + ((size_t)(b * H + hh) * S + qt * 16) * 64;
  const float* Kb = k + ((size_t)(b * H + hh) * S) * 64;
  const float* Vb = v + ((size_t)(b * H + hh) * S) * 64;

  __shared__ float Qs[16 * 68], Ks[16 * 68], Vs[16 * 68], Ps[16 * 17];

#pragma unroll
  for (int j = 0; j < 8; ++j) {              // stage Q tile (16x64)
    int idx = lane + j * 32; int row = idx >> 4, c4 = idx & 15;
    float4 t4 = *(const float4*)(Qb + (size_t)row * 64 + c4 * 4);
    float* dp = Qs + row * 68 + c4 * 4;
    dp[0] = t4.x; dp[1] = t4.y; dp[2] = t4.z; dp[3] = t4.w;
  }
  __syncthreads();

  v8f acc[4];
#pragma unroll
  for (int i = 0; i < 4; ++i) acc[i] = v8zero();
  float mrow[8], lrow[8];
#pragma unroll
  for (int r = 0; r < 8; ++r) { mrow[r] = -3.0e38f; lrow[r] = 0.f; }

  for (int kt = 0; kt < S / 16; ++kt) {
#pragma unroll
    for (int j = 0; j < 8; ++j) {            // stage K,V tiles
      int idx = lane + j * 32; int row = idx >> 4, c4 = idx & 15;
      float4 k4 = *(const float4*)(Kb + (size_t)(kt * 16 + row) * 64 + c4 * 4);
      float4 v4 = *(const float4*)(Vb + (size_t)(kt * 16 + row) * 64 + c4 * 4);
      float* kp = Ks + row * 68 + c4 * 4;
      float* vp = Vs + row * 68 + c4 * 4;
      kp[0] = k4.x; kp[1] = k4.y; kp[2] = k4.z; kp[3] = k4.w;
      vp[0] = v4.x; vp[1] = v4.y; vp[2] = v4.z; vp[3] = v4.w;
    }
    __syncthreads();
    v8f sc = v8zero();
#pragma unroll
    for (int kd = 0; kd < 64; kd += 4) {     // S_tile = Q * K^T
      v2f a;  a[0]  = Qs[l16 * 68 + kd + half * 2];
              a[1]  = Qs[l16 * 68 + kd + half * 2 + 1];
      v2f bb; bb[0] = Ks[l16 * 68 + kd + half * 2];
              bb[1] = Ks[l16 * 68 + kd + half * 2 + 1];
      sc = wmma4(a, bb, sc);
    }
#pragma unroll
    for (int r = 0; r < 8; ++r) {            // online softmax per row
      float sv = sc[r] * 0.125f;             // 1/sqrt(64)
      float tmax = sv;
#pragma unroll
      for (int msk = 1; msk < 16; msk <<= 1) tmax = fmaxf(tmax, __shfl_xor(tmax, msk, 32));
      float nm = fmaxf(mrow[r], tmax);
      float p = expf(sv - nm);
      float rs = p;
#pragma unroll
      for (int msk = 1; msk < 16; msk <<= 1) rs += __shfl_xor(rs, msk, 32);
      float corr = expf(mrow[r] - nm);
      lrow[r] = lrow[r] * corr + rs;
      mrow[r] = nm;
      Ps[(r + 8 * half) * 17 + l16] = p;     // C-layout -> LDS for A-layout reuse
#pragma unroll
      for (int nt = 0; nt < 4; ++nt) acc[nt][r] *= corr;
    }
    __syncthreads();
#pragma unroll
    for (int nt = 0; nt < 4; ++nt) {         // ctx += P * V
#pragma unroll
      for (int kk = 0; kk < 16; kk += 4) {
        v2f a;  a[0]  = Ps[l16 * 17 + kk + half * 2];
                a[1]  = Ps[l16 * 17 + kk + half * 2 + 1];
        v2f bb; bb[0] = Vs[(kk + half * 2) * 68 + nt * 16 + l16];
                bb[1] = Vs[(kk + half * 2 + 1) * 68 + nt * 16 + l16];
        acc[nt] = wmma4(a, bb, acc[nt]);
      }
    }
    __syncthreads();
  }
#pragma unroll
  for (int nt = 0; nt < 4; ++nt) {
#pragma unroll
    for (int r = 0; r < 8; ++r) {
      int row = qt * 16 + r + 8 * half;
      size_t t = (size_t)b * S + row;
      int d = hh * 64 + nt * 16 + l16;
      ctx[t * D + d] = acc[nt][r] / lrow[r];
    }
  }
}

// ---------------- Router: logits, softmax-free top-2 weights ----------------
__global__ __launch_bounds__(256) void gate_kernel(const float* __restrict__ m,
                                                   const float* __restrict__ Wg,
                                                   float* __restrict__ topkw,
                                                   int* __restrict__ topki, int D, int E) {
  const int t = blockIdx.x, tid = threadIdx.x;
  float part[8];
#pragma unroll
  for (int e = 0; e < 8; ++e) part[e] = 0.f;
  const float* mr = m + (size_t)t * D;
  for (int d = tid; d < D; d += 256) {
    float xv = mr[d];
    const float* wr = Wg + (size_t)d * E;
#pragma unroll
    for (int e = 0; e < 8; ++e) part[e] += xv * wr[e];
  }
  __shared__ float red[8][256];
#pragma unroll
  for (int e = 0; e < 8; ++e) red[e][tid] = part[e];
  __syncthreads();
  for (int st = 128; st > 0; st >>= 1) {
    if (tid < st) {
#pragma unroll
      for (int e = 0; e < 8; ++e) red[e][tid] += red[e][tid + st];
    }
    __syncthreads();
  }
  if (tid == 0) {
    float lg[8];
#pragma unroll
    for (int e = 0; e < 8; ++e) lg[e] = red[e][0];
    int i0 = 0; float b0 = lg[0];
    for (int e = 1; e < 8; ++e) if (lg[e] > b0) { b0 = lg[e]; i0 = e; }
    int i1 = (i0 == 0) ? 1 : 0; float b1v = lg[i1];
    for (int e = 0; e < 8; ++e) { if (e == i0) continue; if (lg[e] > b1v) { b1v = lg[e]; i1 = e; } }
    // normalized top-2 softmax weights: w0 = e^l0/(e^l0+e^l1)
    float w0 = 1.f / (1.f + expf(b1v - b0));
    topki[2 * t] = i0; topki[2 * t + 1] = i1;
    topkw[2 * t] = w0; topkw[2 * t + 1] = 1.f - w0;
  }
}

// ---------------- Deterministic expert bucketing (no atomics) ----------------
__global__ __launch_bounds__(256) void count_kernel(const int* __restrict__ topki,
                                                    int* __restrict__ cnt, int T) {
  const int e = blockIdx.x;
  __shared__ int sred[256];
  int c = 0;
  for (int t = threadIdx.x; t < T; t += 256)
    c += (topki[2 * t] == e) + (topki[2 * t + 1] == e);
  sred[threadIdx.x] = c;
  __syncthreads();
  for (int s = 128; s > 0; s >>= 1) {
    if (threadIdx.x < s) sred[threadIdx.x] += sred[threadIdx.x + s];
    __syncthreads();
  }
  if (threadIdx.x == 0) cnt[e] = sred[0];
}

__global__ void prefix_kernel(const int* __restrict__ cnt, int* __restrict__ off) {
  if (threadIdx.x == 0 && blockIdx.x == 0) {
    int a = 0;
    for (int e = 0; e < 8; ++e) { off[e] = a; a += cnt[e]; }
  }
}

__global__ __launch_bounds__(256) void bucket_kernel(const int* __restrict__ topki,
                                                     const int* __restrict__ off,
                                                     int* __restrict__ perm,
                                                     int* __restrict__ slot, int T) {
  const int e = blockIdx.x;
  __shared__ int sc[256];
  __shared__ int sbase;
  if (threadIdx.x == 0) sbase = off[e];
  __syncthreads();
  for (int t0 = 0; t0 < T; t0 += 256) {
    int t = t0 + threadIdx.x;
    int j = -1;
    if (t < T) {
      if (topki[2 * t] == e) j = 0;
      else if (topki[2 * t + 1] == e) j = 1;
    }
    int f = (j >= 0) ? 1 : 0;
    sc[threadIdx.x] = f;
    __syncthreads();
    for (int s = 1; s < 256; s <<= 1) {        // stable inclusive scan
      int val = (threadIdx.x >= s) ? sc[threadIdx.x - s] : 0;
      __syncthreads();
      sc[threadIdx.x] += val;
      __syncthreads();
    }
    int incl = sc[threadIdx.x], total = sc[255];
    if (f) {
      int pos = sbase + incl - 1;
      perm[pos] = t;
      slot[2 * t + j] = pos;
    }
    __syncthreads();
    if (threadIdx.x == 0) sbase += total;
    __syncthreads();
  }
}

// ---------------- Fused grouped-expert FFN: 16 tokens x (D->F gelu->D) ----------------
// Activation rows are gathered into LDS by the Tensor Data Mover in gather mode:
// 16 row indices (16-bit), tile 256 x f32, pad 1 DWORD every 256 DWORDs -> LDS stride 257.
__global__ __launch_bounds__(256) void moe_kernel(const float* __restrict__ m,
                                                  const float* __restrict__ W1,
                                                  const float* __restrict__ b1,
                                                  const float* __restrict__ W2,
                                                  const float* __restrict__ b2,
                                                  const int* __restrict__ cnt,
                                                  const int* __restrict__ off,
                                                  const int* __restrict__ perm,
                                                  float* __restrict__ ye, int D, int F) {
  const int e = blockIdx.y, rt = blockIdx.x;
  const int ce = cnt[e];
  if (rt * 16 >= ce) return;                    // block-uniform exit
  const int base = off[e];
  const int tid = threadIdx.x;
  const int lane = tid & 31, wave = tid >> 5;
  const int half = lane >> 4, l16 = lane & 15;

  __shared__ float As[16 * 257];                // 16 tokens x 256 k-chunk (pad 257)
  __shared__ float Hs[16 * 513];                // 16 tokens x 512 f-chunk (pad 513)
  __shared__ int toks[16];
  if (tid < 16) {
    int srow = rt * 16 + tid;
    // clamp tail rows to the LAST valid token: keeps gather indices non-decreasing
    toks[tid] = perm[base + (srow < ce ? srow : (ce - 1))];
  }
  __syncthreads();

  const unsigned int asOff = lds_offset_of(As);

  v8f acc2[8];
#pragma unroll
  for (int i = 0; i < 8; ++i) acc2[i] = v8zero();

  for (int ch = 0; ch < F; ch += 512) {
    v8f a1[4];
#pragma unroll
    for (int i = 0; i < 4; ++i) a1[i] = v8zero();
    for (int ks = 0; ks < D; ks += 256) {
      __syncthreads();                          // all prior As/Hs consumers done
      if (wave == 0) {
        // --- TDM descriptor (D#), gather mode, 16-bit indices ---
        unsigned long long ga = (unsigned long long)(uintptr_t)(m + ks);
        v4u g0;
        g0.x = 0x80000001u;                     // count=1 | gather_mode | 16-bit idx
        g0.y = asOff;                           // LDS dest byte address
        g0.z = (unsigned int)ga;                // global tile start [31:0]
        g0.w = ((unsigned int)(ga >> 32) & 0x01FFFFFFu) | 0x80000000u; // [56:32] | type=2
        v8i g1;
        g1[0] = (2 << 16)                       // data_size = 4B
              | (1 << 20)                       // pad_enable
              | (7 << 22);                      // pad_interval: every 256 DWORDs (+1 DWORD)
        g1[1] = (int)(256u << 16);              // tensor_dim0 = 256 (lo16 in [31:16])
        g1[2] = (int)(2048u << 16);             // dim0 hi=0 | tensor_dim1 = 2048 lo16
        g1[3] = (int)(256u << 16);              // dim1 hi=0 | tile_dim0 = 256
        g1[4] = 16;                             // tile_dim1 = #valid gather indices
        g1[5] = D;                              // tensor_dim0_stride (elements)
        g1[6] = 0;
        g1[7] = 0;
        v4i g2, g3;
#pragma unroll
        for (int i = 0; i < 4; ++i) {
          g2[i] = (toks[2 * i] & 0xFFFF) | (toks[2 * i + 1] << 16);
          g3[i] = (toks[8 + 2 * i] & 0xFFFF) | (toks[8 + 2 * i + 1] << 16);
        }
        v8i g4;                                 // extra arg on this toolchain (zero-filled,
#pragma unroll
        for (int i = 0; i < 8; ++i) g4[i] = 0;  //  matches probe-verified zero call)
        __builtin_amdgcn_tensor_load_to_lds(g0, g1, g2, g3, g4, 0);
        __builtin_amdgcn_s_wait_tensorcnt(0);   // DMA complete for this wave
      }
      __syncthreads();                          // As visible to all waves
#pragma unroll 2
      for (int kk = 0; kk < 256; kk += 4) {     // h1 += X * W1[e]
        v2f a; a[0] = As[l16 * 257 + kk + half * 2];
               a[1] = As[l16 * 257 + kk + half * 2 + 1];
        const float* w1p = W1 + ((size_t)e * D + ks + kk + half * 2) * F + ch + wave * 64 + l16;
#pragma unroll
        for (int sb = 0; sb < 4; ++sb) {
          v2f bb; bb[0] = w1p[sb * 16]; bb[1] = w1p[sb * 16 + F];
          a1[sb] = wmma4(a, bb, a1[sb]);
        }
      }
    }
#pragma unroll
    for (int sb = 0; sb < 4; ++sb) {            // bias + gelu -> LDS
      int col = wave * 64 + sb * 16 + l16;
      float bb = b1[(size_t)e * F + ch + col];
#pragma unroll
      for (int r = 0; r < 8; ++r)
        Hs[(r + 8 * half) * 513 + col] = gelu_tanh(a1[sb][r] + bb);
    }
    __syncthreads();
#pragma unroll 2
    for (int kk = 0; kk < 512; kk += 4) {       // ye += h1 * W2[e]
      v2f a; a[0] = Hs[l16 * 513 + kk + half * 2];
             a[1] = Hs[l16 * 513 + kk + half * 2 + 1];
      const float* w2p = W2 + ((size_t)e * F + ch + kk + half * 2) * D + wave * 128 + l16;
#pragma unroll
      for (int sb = 0; sb < 8; ++sb) {
        v2f bb; bb[0] = w2p[sb * 16]; bb[1] = w2p[sb * 16 + D];
        acc2[sb] = wmma4(a, bb, acc2[sb]);
      }
    }
  }
#pragma unroll
  for (int sb = 0; sb < 8; ++sb) {
    int d = wave * 128 + sb * 16 + l16;
    float bb = b2[(size_t)e * D + d];
#pragma unroll
    for (int r = 0; r < 8; ++r) {
      int srow = rt * 16 + r + 8 * half;
      if (srow < ce) ye[(size_t)(base + srow) * D + d] = acc2[sb][r] + bb;
    }
  }
}

// ---------------- Final: out = x2 + w0*ye[slot0] + w1*ye[slot1] ----------------
__global__ __launch_bounds__(256) void combine_kernel(const float* __restrict__ x2,
                                                      const float* __restrict__ ye,
                                                      const float* __restrict__ topkw,
                                                      const int* __restrict__ slot,
                                                      float* __restrict__ out, int D) {
  const int t = blockIdx.x, tid = threadIdx.x;
  const float w0 = topkw[2 * t], w1 = topkw[2 * t + 1];
  const size_t s0 = (size_t)slot[2 * t], s1 = (size_t)slot[2 * t + 1];
  const size_t bo = (size_t)t * D;
  for (int d = tid; d < D; d += 256)
    out[bo + d] = x2[bo + d] + w0 * ye[s0 * D + d] + w1 * ye[s1 * D + d];
}

extern "C" void kernel_launch(void* const* d_in, const int* in_sizes, int n_in,
                              void* d_out, int out_size, void* d_ws, size_t ws_size,
                              hipStream_t stream) {
  (void)in_sizes; (void)n_in; (void)out_size; (void)ws_size;
  const float* x    = (const float*)d_in[0];
  const float* ln1g = (const float*)d_in[1];
  const float* ln1b = (const float*)d_in[2];
  const float* ln2g = (const float*)d_in[3];
  const float* ln2b = (const float*)d_in[4];
  const float* Wq   = (const float*)d_in[5];
  const float* Wk   = (const float*)d_in[6];
  const float* Wv   = (const float*)d_in[7];
  const float* Wo   = (const float*)d_in[8];
  const float* Wg   = (const float*)d_in[9];
  const float* W1   = (const float*)d_in[10];
  const float* b1   = (const float*)d_in[11];
  const float* W2   = (const float*)d_in[12];
  const float* b2   = (const float*)d_in[13];

  const int Bn = 2, S = 1024, D = 1024, H = 16, Dh = 64, E = 8, F = 4096;
  const int T = Bn * S;
  const size_t TD = (size_t)T * D;

  float* ws   = (float*)d_ws;
  float* h    = ws;                 // LN1 out; reused as LN2 out (m)
  float* qb   = ws + TD;            // reused as ye (spans q+k regions: 2*TD)
  float* kb   = ws + 2 * TD;
  float* vb   = ws + 3 * TD;
  float* ctx  = ws + 4 * TD;
  float* x2   = ws + 5 * TD;
  float* mbuf = h;
  float* ye   = qb;
  float* topkw = ws + 6 * TD;                       // 2T floats
  int*   topki = (int*)(ws + 6 * TD + 2 * T);       // 2T ints
  int*   cnts  = topki + 2 * T;                     // E
  int*   offs  = cnts + 8;                          // E
  int*   perm  = offs + 8;                          // 2T
  int*   slot  = perm + 2 * T;                      // 2T
  float* outp  = (float*)d_out;

  ln_kernel<<<T, 256, 0, stream>>>(x, ln1g, ln1b, h, D);
  dim3 gg(D / 64, T / 64);
  gemm64_kernel<1><<<gg, 256, 0, stream>>>(h, Wq, nullptr, qb, T, D, D, S, H, Dh);
  gemm64_kernel<1><<<gg, 256, 0, stream>>>(h, Wk, nullptr, kb, T, D, D, S, H, Dh);
  gemm64_kernel<1><<<gg, 256, 0, stream>>>(h, Wv, nullptr, vb, T, D, D, S, H, Dh);
  attn_kernel<<<dim3(S / 16, H, Bn), 32, 0, stream>>>(qb, kb, vb, ctx, S, H, D);
  gemm64_kernel<2><<<gg, 256, 0, stream>>>(ctx, Wo, x, x2, T, D, D, S, H, Dh);
  ln_kernel<<<T, 256, 0, stream>>>(x2, ln2g, ln2b, mbuf, D);
  gate_kernel<<<T, 256, 0, stream>>>(mbuf, Wg, topkw, topki, D, E);
  count_kernel<<<E, 256, 0, stream>>>(topki, cnts, T);
  prefix_kernel<<<1, 32, 0, stream>>>(cnts, offs);
  bucket_kernel<<<E, 256, 0, stream>>>(topki, offs, perm, slot, T);
  moe_kernel<<<dim3(T / 16, E), 256, 0, stream>>>(mbuf, W1, b1, W2, b2, cnts, offs, perm, ye, D, F);
  combine_kernel<<<T, 256, 0, stream>>>(x2, ye, topkw, slot, outp, D);
}